// AffineTransformLayer_34995393528554
// MI455X (gfx1250) — compile-verified
//
#include <hip/hip_runtime.h>

// AffineTransformLayer: projective warp, nearest-neighbor gather.
// image: (B=16, H=128, W=128, C=64) f32
// theta: (B=16, A=10, 8) f32
// out:   (B, H, W, A*C) f32
//
// HBM-store-bound (671 MB out stream; 67 MB image L2-resident, ~10x reuse).
// Phase 1: 128 threads compute per-pixel src offsets (one div pair per pixel,
//          not per lane) into an LDS table.
// Phase 2: gather via gfx1250 async DMA (global_load_async_to_lds_b128,
//          ASYNCcnt-pipelined), then NT b128 stores from the LDS tile.

typedef float v4f __attribute__((ext_vector_type(4)));

#define BB 16
#define HH 128
#define WW 128
#define CC 64
#define AA 10
#define NTHREADS 256
#define XTILES (WW / 16)   // 8

__global__ __launch_bounds__(NTHREADS) void warp_gather_async_kernel(
    const float* __restrict__ image,
    const float* __restrict__ theta,
    float* __restrict__ out)
{
    __shared__ int s_off[WW];                 // bit31 = invalid, [30:0] = byte offset
    __shared__ v4f s_tile[NTHREADS * XTILES]; // 32 KB staging tile

    // Block decomposition: blk = ((b*A + a)*H + y)
    const int blk = blockIdx.x;
    const int y  = blk % HH;
    const int ba = blk / HH;
    const int a  = ba % AA;
    const int b  = ba / AA;
    const int tid = threadIdx.x;

    // ---- Phase 1: one thread per output pixel computes the source offset ----
    if (tid < WW) {
        const float* __restrict__ t = theta + (size_t)(b * AA + a) * 8; // uniform -> s_load
        const float ta0 = t[0], ta1 = t[1], ta2 = t[2];
        const float tb0 = t[3], tb1 = t[4], tb2 = t[5];
        const float tc0 = t[6], tc1 = t[7];
        const float fy = (float)y;
        const float fx = (float)tid;

        const float k   = tc0 * fx + (tc1 * fy + 1.0f);
        // True divisions: match reference rounding at floor(x+0.5) edges.
        const float xin = (ta0 * fx + (ta1 * fy + ta2)) / k;
        const float yin = (tb0 * fx + (tb1 * fy + tb2)) / k;
        // v_cvt_i32_f32 clamps +/-huge, NaN->0: huge lands outside [0,128) => invalid; NaN->0 passes as valid.
        const int xr = (int)floorf(xin + 0.5f);
        const int yr = (int)floorf(yin + 0.5f);
        const bool valid = (xr >= 0) & (xr < WW) & (yr >= 0) & (yr < HH);
        const int xc = min(max(xr, 0), WW - 1);
        const int yc = min(max(yr, 0), HH - 1);
        const int off = (yc * WW + xc) * (CC * 4); // clamped byte offset in image[b]
        s_off[tid] = valid ? off : (off | (int)0x80000000);
    }
    __syncthreads();

    // ---- Phase 2: async DMA gather into LDS, then NT stores ----
    const int c4 = tid & 15;   // channel group (float4)
    const int ps = tid >> 4;   // pixel-sub within 16-pixel tile
    const unsigned long long imgB =
        (unsigned long long)(size_t)(image + (size_t)b * (HH * WW * CC))
        + (unsigned)(c4 * 16);

    int packed[XTILES];

    #pragma unroll
    for (int xt = 0; xt < XTILES; ++xt) {
        const int x = xt * 16 + ps;
        const int p = s_off[x];                 // LDS broadcast across 16 lanes
        packed[xt] = p;
        const unsigned long long gaddr = imgB + (unsigned)(p & 0x7FFFFFFF);
        // Low 32 bits of a generic pointer to LDS == DS byte offset.
        const unsigned lds_dst = (unsigned)(size_t)(&s_tile[xt * NTHREADS + tid]);
        // gfx1250 async copy: HBM/L2 -> LDS, no VGPR staging, ASYNCcnt-tracked.
        asm volatile("global_load_async_to_lds_b128 %0, %1, off"
                     :: "v"(lds_dst), "v"(gaddr) : "memory");
    }

    float* __restrict__ outRow =
        out + (size_t)(b * HH + y) * WW * (AA * CC) + a * CC + c4 * 4;

    #pragma unroll
    for (int xt = 0; xt < XTILES; ++xt) {
        // Async loads complete in order: drain one tile per iteration.
        asm volatile("s_wait_asynccnt %0" :: "i"(XTILES - 1 - xt) : "memory");
        v4f v = s_tile[xt * NTHREADS + tid];
        if (packed[xt] < 0) v = (v4f){0.0f, 0.0f, 0.0f, 0.0f}; // invalid -> zero
        const int x = xt * 16 + ps;
        // Streaming 671 MB: TH=NT so the output doesn't evict the image from L2.
        __builtin_nontemporal_store(v, (v4f*)(outRow + (size_t)x * (AA * CC)));
    }
}

extern "C" void kernel_launch(void* const* d_in, const int* in_sizes, int n_in,
                              void* d_out, int out_size, void* d_ws, size_t ws_size,
                              hipStream_t stream)
{
    const float* image = (const float*)d_in[0];
    const float* theta = (const float*)d_in[1];
    float* out = (float*)d_out;

    // One block per (b, a, y) output row: 16*10*128 = 20480 blocks.
    dim3 grid(BB * AA * HH);
    dim3 block(NTHREADS);
    warp_gather_async_kernel<<<grid, block, 0, stream>>>(image, theta, out);
}